// AttnDecoderRNN3_40767829574386
// MI455X (gfx1250) — compile-verified
//
#include <hip/hip_runtime.h>
#include <cstddef>

// ---------------------------------------------------------------------------
// MI455X / gfx1250, wave32. fp32 pipeline (memory-bound: ~120MB weights,
// ~2 GFLOP), matrix ops via V_WMMA_F32_16X16X4_F32.
// ---------------------------------------------------------------------------

typedef __attribute__((ext_vector_type(2))) float v2f;
typedef __attribute__((ext_vector_type(8))) float v8f;

#define DEVFN __device__ __forceinline__

DEVFN float sigmoidf_(float x) { return 1.0f / (1.0f + __expf(-x)); }

// ---------------------------------------------------------------------------
// Generic WMMA GEMM:  C[32, N] = act( A[32, K] * W[N, K]^T + bias[N] )
// A, W row-major; K % 4 == 0; N % 16 == 0.
// Block = 256 threads = 8 waves. Wave w: M-tile = w&1, N-tile = blk*4 + w>>1.
// Fragment layouts per CDNA5 ISA 7.12.2:
//   A 16x4 : lane<16 -> {K=k0,k0+1}, lane>=16 -> {K=k0+2,k0+3}; row = lane&15
//   B 4x16 : same K split per lane half;                       col = lane&15
//   C 16x16: vgpr i -> M = i + 8*(lane>=16), N = lane&15
// ---------------------------------------------------------------------------
template <int ACT> // 0 = none, 1 = relu
__global__ __launch_bounds__(256) void wmma_gemm_nt(
    const float* __restrict__ A, const float* __restrict__ W,
    const float* __restrict__ bias, float* __restrict__ C, int N, int K)
{
    const int lane = threadIdx.x & 31;
    const int wave = threadIdx.x >> 5;
    const int hi   = lane >> 4;     // 0 | 1 : which K-pair / which M-half of C
    const int lo   = lane & 15;     // A-row / B-col / C-col within tile
    const int m0   = (wave & 1) << 4;
    const int n0   = (blockIdx.x * 4 + (wave >> 1)) << 4;
    if (n0 >= N) return;            // wave-uniform: EXEC stays all-ones for WMMA

    const float* __restrict__ ap = A + (size_t)(m0 + lo) * K + 2 * hi;
    const float* __restrict__ bp = W + (size_t)(n0 + lo) * K + 2 * hi;

    v8f acc = {};
    int k = 0;
    // Main loop: 64-wide K chunks, prefetch next chunk of the weight stream.
    for (; k + 64 <= K; k += 64) {
        __builtin_prefetch(bp + k + 64, 0, 1);   // -> global_prefetch_b8
#pragma unroll
        for (int kk = 0; kk < 64; kk += 4) {
            v2f a = *(const v2f*)(ap + k + kk);
            v2f b = *(const v2f*)(bp + k + kk);
            acc = __builtin_amdgcn_wmma_f32_16x16x4_f32(
                false, a, false, b, (short)0, acc, false, false);
        }
    }
    for (; k < K; k += 4) {
        v2f a = *(const v2f*)(ap + k);
        v2f b = *(const v2f*)(bp + k);
        acc = __builtin_amdgcn_wmma_f32_16x16x4_f32(
            false, a, false, b, (short)0, acc, false, false);
    }

#pragma unroll
    for (int i = 0; i < 8; ++i) {
        int m = m0 + i + 8 * hi;
        int n = n0 + lo;
        float v = acc[i];
        if (bias) v += bias[n];
        if (ACT == 1) v = fmaxf(v, 0.0f);
        C[(size_t)m * N + n] = v;
    }
}

// ---------------------------------------------------------------------------
// Speed MLP: sp[n,c] = tanh( relu(speed[n]*spd1_w + spd1_b) @ spd2_w.T + spd2_b )
// ---------------------------------------------------------------------------
__global__ __launch_bounds__(256) void speed_mlp_kernel(
    const float* __restrict__ speed, const float* __restrict__ spd1_w,
    const float* __restrict__ spd1_b, const float* __restrict__ spd2_w,
    const float* __restrict__ spd2_b, float* __restrict__ sp)
{
    __shared__ float hid[512];
    const int n = blockIdx.x, t = threadIdx.x;
    const float s = speed[n];
    hid[t]       = fmaxf(s * spd1_w[t]       + spd1_b[t],       0.0f);
    hid[t + 256] = fmaxf(s * spd1_w[t + 256] + spd1_b[t + 256], 0.0f);
    __syncthreads();
    for (int c = t; c < 512; c += 256) {
        const float* wr = spd2_w + (size_t)c * 512;
        float acc = spd2_b[c];
        for (int k = 0; k < 512; ++k) acc += hid[k] * wr[k];
        sp[(size_t)n * 512 + c] = tanhf(acc);
    }
}

DEVFN float block_reduce_sum(float v, float* red)
{
    const int t = threadIdx.x;
    red[t] = v; __syncthreads();
    for (int s = 128; s > 0; s >>= 1) {
        if (t < s) red[t] += red[t + s];
        __syncthreads();
    }
    float r = red[0];
    __syncthreads();
    return r;
}

// ---------------------------------------------------------------------------
// Collapsed attention + context. One block per batch n, 256 threads = ATT dim.
// Only t in [0, ATT_RANGE) can have nonzero weight (aw0 one-hot at t=0); the
// global-max shift in the reference softmax cancels exactly, so a local max
// over the masked window is mathematically identical.
// ---------------------------------------------------------------------------
__global__ __launch_bounds__(256) void attention_context_kernel(
    const float* __restrict__ input_enc, const float* __restrict__ spkr_vec,
    const float* __restrict__ speed, const int* __restrict__ lengths_enc,
    const float* __restrict__ enc_w, const float* __restrict__ enc_b,
    const float* __restrict__ spkr_w, const float* __restrict__ conv_w,
    const float* __restrict__ speed_att_w, const float* __restrict__ attproj_w,
    const float* __restrict__ attproj_b, const float* __restrict__ sp,
    float* __restrict__ ctx_ws, float* __restrict__ ctx_out)
{
    __shared__ float red[256];
    __shared__ float logits[10];
    __shared__ float wgt[10];
    const int n = blockIdx.x;
    const int a = threadIdx.x;   // 256 == ATT

    // speaker bias: softsign(spkr_vec @ spkr_w.T)
    const float* sv  = spkr_vec + (size_t)n * 64;
    const float* swr = spkr_w + (size_t)a * 64;
    float sb = 0.0f;
    for (int k = 0; k < 64; ++k) sb += sv[k] * swr[k];
    sb = sb / (1.0f + fabsf(sb));
    const float spd_t = speed[n] * speed_att_w[a];
    const float* ew = enc_w + (size_t)a * 512;

    for (int t = 0; t < 10; ++t) {
        const float* er = input_enc + ((size_t)n * 2048 + t) * 512;
        float d = 0.0f;
        for (int k = 0; k < 512; ++k) d += er[k] * ew[k];
        float e = d + enc_b[a];
        e = e / (1.0f + fabsf(e));                       // softsign
        e += sb + conv_w[a * 31 + (15 - t)] + spd_t;     // conv of one-hot
        float partial = tanhf(e) * attproj_w[a];
        float s = block_reduce_sum(partial, red);
        if (a == 0) logits[t] = s + attproj_b[0];
        __syncthreads();
    }

    if (a == 0) {
        const int lm1 = lengths_enc[n] - 1;
        float mx = -3.0e38f;
        for (int t = 0; t < 10; ++t) if (t <= lm1) mx = fmaxf(mx, logits[t]);
        float s = 0.0f;
        for (int t = 0; t < 10; ++t) {
            float w = (t <= lm1) ? __expf(logits[t] - mx) : 0.0f;
            wgt[t] = w; s += w;
        }
        s = fmaxf(s, 1e-12f);
        for (int t = 0; t < 10; ++t) wgt[t] /= s;
    }
    __syncthreads();

    // context[n,c] = sum_t w[t] * (input_enc[n,t,c] + sp[n,c])
    const float* spn = sp + (size_t)n * 512;
    for (int c = a; c < 512; c += 256) {
        float acc = 0.0f;
        for (int t = 0; t < 10; ++t)
            acc += wgt[t] * (input_enc[((size_t)n * 2048 + t) * 512 + c] + spn[c]);
        ctx_ws[(size_t)n * 512 + c] = acc;
        ctx_out[(size_t)n * 512 + c] = acc;
    }
}

// ---------------------------------------------------------------------------
// Small glue kernels
// ---------------------------------------------------------------------------
__global__ void build_pin_kernel(const float* __restrict__ input_dec,
                                 const float* __restrict__ spkr_vec,
                                 float* __restrict__ pin)
{
    const int n = blockIdx.x, j = threadIdx.x;   // 144 threads
    pin[n * 144 + j] = (j < 80) ? input_dec[n * 80 + j]
                                : spkr_vec[n * 64 + (j - 80)];
}

__global__ void build_inlstm_kernel(const float* __restrict__ pre,
                                    const float* __restrict__ ctx,
                                    const float* __restrict__ spkr,
                                    float* __restrict__ out)
{
    const int n = blockIdx.x;
    for (int j = threadIdx.x; j < 1088; j += blockDim.x) {
        float v = (j < 512) ? pre[n * 512 + j]
                : (j < 1024) ? ctx[n * 512 + (j - 512)]
                             : spkr[n * 64 + (j - 1024)];
        out[n * 1088 + j] = v;
    }
}

__global__ void build_xcat_kernel(const float* __restrict__ h2,
                                  const float* __restrict__ inl,
                                  float* __restrict__ out)
{
    const int n = blockIdx.x;
    for (int j = threadIdx.x; j < 3136; j += blockDim.x)
        out[n * 3136 + j] = (j < 2048) ? h2[n * 2048 + j]
                                       : inl[n * 1088 + (j - 2048)];
}

// gates: g split [i | f | gg | o] (2048 each); c=sig(i)*tanh(gg); h=sig(o)*tanh(c)
__global__ __launch_bounds__(256) void lstm_gates_kernel(
    const float* __restrict__ graw, const float* __restrict__ bih,
    const float* __restrict__ bhh, float* __restrict__ h)
{
    const int idx = blockIdx.x * 256 + threadIdx.x;   // 32*2048
    const int n = idx >> 11, j = idx & 2047;
    const float* gr = graw + (size_t)n * 8192;
    const float gi = gr[j]        + bih[j]        + bhh[j];
    const float gg = gr[4096 + j] + bih[4096 + j] + bhh[4096 + j];
    const float go = gr[6144 + j] + bih[6144 + j] + bhh[6144 + j];
    const float c  = sigmoidf_(gi) * tanhf(gg);
    h[idx] = sigmoidf_(go) * tanhf(c);
}

// out = LN(x + r) * g + b, 512 features per row, in-place-safe (out may == x)
__global__ __launch_bounds__(256) void add_layernorm_kernel(
    const float* __restrict__ x, const float* __restrict__ r,
    const float* __restrict__ g, const float* __restrict__ b,
    float* __restrict__ out)
{
    __shared__ float red[256];
    const int n = blockIdx.x, t = threadIdx.x;
    const float v0 = x[n * 512 + t]       + r[n * 512 + t];
    const float v1 = x[n * 512 + 256 + t] + r[n * 512 + 256 + t];
    const float mean = block_reduce_sum(v0 + v1, red) * (1.0f / 512.0f);
    const float d0 = v0 - mean, d1 = v1 - mean;
    const float var = block_reduce_sum(d0 * d0 + d1 * d1, red) * (1.0f / 512.0f);
    const float inv = rsqrtf(var + 1e-5f);
    out[n * 512 + t]       = d0 * inv * g[t]       + b[t];
    out[n * 512 + 256 + t] = d1 * inv * g[t + 256] + b[t + 256];
}

// ---------------------------------------------------------------------------
extern "C" void kernel_launch(void* const* d_in, const int* in_sizes, int n_in,
                              void* d_out, int out_size, void* d_ws, size_t ws_size,
                              hipStream_t stream)
{
    const float* input_enc   = (const float*)d_in[0];
    const float* input_dec   = (const float*)d_in[1];
    const float* spkr_vec    = (const float*)d_in[2];
    const float* speed       = (const float*)d_in[3];
    const int*   lengths_enc = (const int*)  d_in[4];
    const float* enc_w       = (const float*)d_in[5];
    const float* enc_b       = (const float*)d_in[6];
    const float* spkr_w      = (const float*)d_in[7];
    const float* conv_w      = (const float*)d_in[8];
    const float* speed_att_w = (const float*)d_in[9];
    const float* attproj_w   = (const float*)d_in[10];
    const float* attproj_b   = (const float*)d_in[11];
    const float* spd1_w      = (const float*)d_in[12];
    const float* spd1_b      = (const float*)d_in[13];
    const float* spd2_w      = (const float*)d_in[14];
    const float* spd2_b      = (const float*)d_in[15];
    const float* pre1_w      = (const float*)d_in[16];
    const float* pre1_b      = (const float*)d_in[17];
    const float* pre2_w      = (const float*)d_in[18];
    const float* pre2_b      = (const float*)d_in[19];
    const float* lstm_wih0   = (const float*)d_in[20];
    const float* lstm_bih0   = (const float*)d_in[22];
    const float* lstm_bhh0   = (const float*)d_in[23];
    const float* lstm_wih1   = (const float*)d_in[24];
    const float* lstm_bih1   = (const float*)d_in[26];
    const float* lstm_bhh1   = (const float*)d_in[27];
    const float* outl1_w     = (const float*)d_in[28];
    const float* outl1_b     = (const float*)d_in[29];
    const float* tr_inproj_w = (const float*)d_in[30];
    const float* tr_inproj_b = (const float*)d_in[31];
    const float* tr_out_w    = (const float*)d_in[32];
    const float* tr_out_b    = (const float*)d_in[33];
    const float* tr_ln1_g    = (const float*)d_in[34];
    const float* tr_ln1_b    = (const float*)d_in[35];
    const float* tr_ff1_w    = (const float*)d_in[36];
    const float* tr_ff1_b    = (const float*)d_in[37];
    const float* tr_ff2_w    = (const float*)d_in[38];
    const float* tr_ff2_b    = (const float*)d_in[39];
    const float* tr_ln2_g    = (const float*)d_in[40];
    const float* tr_ln2_b    = (const float*)d_in[41];
    const float* outl2_w     = (const float*)d_in[42];
    const float* outl2_b     = (const float*)d_in[43];

    float* out_main = (float*)d_out;            // 32*160 floats
    float* out_ctx  = (float*)d_out + 32 * 160; // 32*512 floats

    // workspace layout (floats), ~3.7 MB total
    float* f     = (float*)d_ws;
    float* sp    = f;               f += 32 * 512;
    float* ctx   = f;               f += 32 * 512;
    float* pin   = f;               f += 32 * 144;
    float* preh  = f;               f += 32 * 1024;
    float* pre   = f;               f += 32 * 512;
    float* inl   = f;               f += 32 * 1088;
    float* graw  = f;               f += 32 * 8192;
    float* h1    = f;               f += 32 * 2048;
    float* graw2 = f;               f += 32 * 8192;
    float* h2    = f;               f += 32 * 2048;
    float* xcat  = f;               f += 32 * 3136;
    float* xbuf  = f;               f += 32 * 512;
    float* vbuf  = f;               f += 32 * 512;
    float* attb  = f;               f += 32 * 512;
    float* ffh   = f;               f += 32 * 1024;
    float* ffo   = f;               f += 32 * 512;

    // 1. speed MLP -> sp (32,512)
    speed_mlp_kernel<<<32, 256, 0, stream>>>(speed, spd1_w, spd1_b, spd2_w, spd2_b, sp);
    // 2. collapsed attention + context (writes ws copy AND d_out tail)
    attention_context_kernel<<<32, 256, 0, stream>>>(
        input_enc, spkr_vec, speed, lengths_enc, enc_w, enc_b, spkr_w, conv_w,
        speed_att_w, attproj_w, attproj_b, sp, ctx, out_ctx);
    // 3. pre-net
    build_pin_kernel<<<32, 144, 0, stream>>>(input_dec, spkr_vec, pin);
    wmma_gemm_nt<1><<<16, 256, 0, stream>>>(pin,  pre1_w, pre1_b, preh, 1024, 144);
    wmma_gemm_nt<1><<<8,  256, 0, stream>>>(preh, pre2_w, pre2_b, pre,  512, 1024);
    // 4. LSTM input = [pre | context | spkr]
    build_inlstm_kernel<<<32, 256, 0, stream>>>(pre, ctx, spkr_vec, inl);
    // 5. "LSTM" layers (no recurrence; whh unused by the reference)
    wmma_gemm_nt<0><<<128, 256, 0, stream>>>(inl, lstm_wih0, nullptr, graw, 8192, 1088);
    lstm_gates_kernel<<<256, 256, 0, stream>>>(graw, lstm_bih0, lstm_bhh0, h1);
    wmma_gemm_nt<0><<<128, 256, 0, stream>>>(h1, lstm_wih1, nullptr, graw2, 8192, 2048);
    lstm_gates_kernel<<<256, 256, 0, stream>>>(graw2, lstm_bih1, lstm_bhh1, h2);
    // 6. x = [h2 | in_lstm] @ outl1
    build_xcat_kernel<<<32, 256, 0, stream>>>(h2, inl, xcat);
    wmma_gemm_nt<0><<<8, 256, 0, stream>>>(xcat, outl1_w, outl1_b, xbuf, 512, 3136);
    // 7. transformer, S=1 => attention output == V projection exactly
    for (int l = 0; l < 2; ++l) {
        const float* wv  = tr_inproj_w + (size_t)l * 1536 * 512 + (size_t)1024 * 512;
        const float* bv  = tr_inproj_b + (size_t)l * 1536 + 1024;
        const float* wo  = tr_out_w + (size_t)l * 512 * 512;
        const float* bo  = tr_out_b + (size_t)l * 512;
        wmma_gemm_nt<0><<<8, 256, 0, stream>>>(xbuf, wv, bv, vbuf, 512, 512);
        wmma_gemm_nt<0><<<8, 256, 0, stream>>>(vbuf, wo, bo, attb, 512, 512);
        add_layernorm_kernel<<<32, 256, 0, stream>>>(
            xbuf, attb, tr_ln1_g + l * 512, tr_ln1_b + l * 512, xbuf);
        wmma_gemm_nt<1><<<16, 256, 0, stream>>>(
            xbuf, tr_ff1_w + (size_t)l * 1024 * 512, tr_ff1_b + l * 1024, ffh, 1024, 512);
        wmma_gemm_nt<0><<<8, 256, 0, stream>>>(
            ffh, tr_ff2_w + (size_t)l * 512 * 1024, tr_ff2_b + l * 512, ffo, 512, 1024);
        add_layernorm_kernel<<<32, 256, 0, stream>>>(
            xbuf, ffo, tr_ln2_g + l * 512, tr_ln2_b + l * 512, xbuf);
    }
    // 8. output head -> d_out[0 : 32*160]  (row-major (n, 160) matches
    //    (1,32,160).transpose(1,0,2).reshape(32,2,80) flat layout)
    wmma_gemm_nt<0><<<3, 256, 0, stream>>>(xbuf, outl2_w, outl2_b, out_main, 160, 512);

    (void)in_sizes; (void)n_in; (void)out_size; (void)ws_size;
}